// dgl_gat_73529840107892
// MI455X (gfx1250) — compile-verified
//
#include <hip/hip_runtime.h>
#include <math.h>

typedef __attribute__((ext_vector_type(2))) float v2f;
typedef __attribute__((ext_vector_type(8))) float v8f;

#define IN_DIM 256
#define NHEAD1 4
#define DIM1   32
#define HD1    128
#define NCLASS 16

// ---------------------------------------------------------------------------
// LDS-staged fp32 GEMM using V_WMMA_F32_16X16X4_F32.
// C[M,Ncols] = A[M,K] @ W[Ncols,K]^T   (W row-major [out,in])
// One workgroup = NT waves handles a 16-row stripe and all NT*16 columns.
// The 16xK A stripe is staged once in LDS (coalesced float4 -> ds_store_b128),
// then each wave re-reads its per-lane WMMA fragments via ds_load_b64.
// Fragment layout (16x16x4 f32):
//   A: lanes 0-15 row M=lane K={k,k+1}; lanes 16-31 same rows K={k+2,k+3}
//   B: mirrored with N = lane&15;   B[k][n] = W[n][k]
//   C/D: VGPR i -> M = i + 8*(lane>=16), N = lane&15
// ---------------------------------------------------------------------------
template <int K, int NT>
__global__ __launch_bounds__(32 * NT) void k_gemm_wmma_lds(
    const float* __restrict__ A, const float* __restrict__ W,
    float* __restrict__ C, int M, int Ncols) {
  __shared__ float As[16 * K];

  const int tid  = threadIdx.x;
  const int row0 = blockIdx.x * 16;

  // ---- stage A stripe (16 rows x K) into LDS, coalesced float4 ----
  constexpr int TOT4 = (16 * K) / 4;
  for (int i = tid; i < TOT4; i += 32 * NT) {
    const int elem = i * 4;
    const int row  = elem / K;
    const int col  = elem - row * K;
    const int grow = min(row0 + row, M - 1);   // clamp: keep all lanes valid
    const float4 v = *(const float4*)(A + (size_t)grow * K + col);
    *(float4*)(As + elem) = v;
  }
  __syncthreads();

  const int wave = tid >> 5;
  const int lane = tid & 31;
  const int half = lane >> 4;
  const int r    = lane & 15;
  const int col0 = wave * 16;

  const float* __restrict__ arow = As + r * K;                 // LDS
  const float* __restrict__ wrow = W + (size_t)(col0 + r) * K; // global (L2-hot)

  v8f acc = {};
#pragma unroll 8
  for (int k0 = 0; k0 < K; k0 += 4) {
    const int k = k0 + 2 * half;
    v2f a; a.x = arow[k]; a.y = arow[k + 1];   // ds_load_b64
    v2f b; b.x = wrow[k]; b.y = wrow[k + 1];   // global_load_b64
    acc = __builtin_amdgcn_wmma_f32_16x16x4_f32(
        false, a, false, b, (short)0, acc, false, false);
  }

#pragma unroll
  for (int i = 0; i < 8; ++i) {
    const int m = row0 + i + 8 * half;
    if (m < M) C[(size_t)m * Ncols + (col0 + r)] = acc[i];
  }
}

// ---------------------------------------------------------------------------
__global__ void k_fill(float* __restrict__ p, int n, float v) {
  const int i = blockIdx.x * blockDim.x + threadIdx.x;
  if (i < n) p[i] = v;
}

// el[n,h] = sum_d h[n,h,d]*attn_l[h,d] ; er likewise
__global__ void k_el_er(const float* __restrict__ h,
                        const float* __restrict__ al,
                        const float* __restrict__ ar,
                        float* __restrict__ el, float* __restrict__ er,
                        int n_nodes, int H, int D) {
  const int idx = blockIdx.x * blockDim.x + threadIdx.x;
  if (idx >= n_nodes * H) return;
  const int n = idx / H, hh = idx - n * H;
  const float* __restrict__ hp = h + ((size_t)n * H + hh) * D;
  const float* __restrict__ lp = al + hh * D;
  const float* __restrict__ rp = ar + hh * D;
  float sl = 0.f, sr = 0.f;
  for (int d = 0; d < D; ++d) { const float v = hp[d]; sl += v * lp[d]; sr += v * rp[d]; }
  el[idx] = sl;
  er[idx] = sr;
}

// segment max of edge scores into m[dst,h]  (global_atomic_max_num_f32)
__global__ void k_edge_max(const int* __restrict__ src, const int* __restrict__ dst,
                           const float* __restrict__ el, const float* __restrict__ er,
                           float* __restrict__ m, int n_edges, int H) {
  const int idx = blockIdx.x * blockDim.x + threadIdx.x;
  if (idx >= n_edges * H) return;
  const int e = idx / H, hh = idx - e * H;
  const int s = src[e], d = dst[e];
  atomicMax(&m[d * H + hh], el[s * H + hh] + er[d * H + hh]);
}

// z[dst,h] += exp(e - m[dst,h])
__global__ void k_edge_expsum(const int* __restrict__ src, const int* __restrict__ dst,
                              const float* __restrict__ el, const float* __restrict__ er,
                              const float* __restrict__ m, float* __restrict__ z,
                              int n_edges, int H) {
  const int idx = blockIdx.x * blockDim.x + threadIdx.x;
  if (idx >= n_edges * H) return;
  const int e = idx / H, hh = idx - e * H;
  const int s = src[e], d = dst[e];
  const float v = el[s * H + hh] + er[d * H + hh] - m[d * H + hh];
  atomicAdd(&z[d * H + hh], expf(v));
}

// out[dst, j0..j0+3] += alpha(e,h) * h[src, j0..j0+3]
// one thread per float4 chunk; alpha recomputed (head uniform per chunk
// since D % 4 == 0). h gather is a global_load_b128.
__global__ void k_aggregate4(const int* __restrict__ src, const int* __restrict__ dst,
                             const float* __restrict__ el, const float* __restrict__ er,
                             const float* __restrict__ m, const float* __restrict__ z,
                             const float* __restrict__ h, float* __restrict__ out,
                             int n_edges, int H, int D) {
  const int HD = H * D;
  const int nv = HD >> 2;  // float4 chunks per edge
  const long long idx = (long long)blockIdx.x * blockDim.x + threadIdx.x;
  if (idx >= (long long)n_edges * nv) return;
  const int e  = (int)(idx / nv);
  const int q  = (int)(idx - (long long)e * nv);
  const int j0 = q << 2;
  const int hh = j0 / D;
  const int s = src[e], d = dst[e];
  const float v = el[s * H + hh] + er[d * H + hh] - m[d * H + hh];
  const float alpha = expf(v) / z[d * H + hh];
  const float4 hv = *(const float4*)(h + (size_t)s * HD + j0);
  float* op = out + (size_t)d * HD + j0;
  atomicAdd(op + 0, alpha * hv.x);
  atomicAdd(op + 1, alpha * hv.y);
  atomicAdd(op + 2, alpha * hv.z);
  atomicAdd(op + 3, alpha * hv.w);
}

// x = elu(x + bias[j % F]), in place
__global__ void k_bias_elu(float* __restrict__ x, const float* __restrict__ bias,
                           int total, int F) {
  const int idx = blockIdx.x * blockDim.x + threadIdx.x;
  if (idx >= total) return;
  const float v = x[idx] + bias[idx % F];
  x[idx] = v > 0.f ? v : expm1f(v);
}

// ---------------------------------------------------------------------------
extern "C" void kernel_launch(void* const* d_in, const int* in_sizes, int n_in,
                              void* d_out, int out_size, void* d_ws, size_t ws_size,
                              hipStream_t stream) {
  const float* features = (const float*)d_in[0];
  const int*   src      = (const int*)d_in[1];
  const int*   dst      = (const int*)d_in[2];
  const float* fc1_w    = (const float*)d_in[3];
  const float* attn_l1  = (const float*)d_in[4];
  const float* attn_r1  = (const float*)d_in[5];
  const float* bias1    = (const float*)d_in[6];
  const float* fc2_w    = (const float*)d_in[7];
  const float* attn_l2  = (const float*)d_in[8];
  const float* attn_r2  = (const float*)d_in[9];
  const float* bias2    = (const float*)d_in[10];
  float* out = (float*)d_out;

  const int n_nodes = in_sizes[0] / IN_DIM;
  const int n_edges = in_sizes[1];

  // workspace layout (floats)
  float* ws  = (float*)d_ws;
  float* h1  = ws;                                  // 128N
  float* x1  = h1  + (size_t)HD1 * n_nodes;         // 128N (agg1, then ELU'd x)
  float* el1 = x1  + (size_t)HD1 * n_nodes;         // 4N
  float* er1 = el1 + (size_t)NHEAD1 * n_nodes;      // 4N
  float* m1  = er1 + (size_t)NHEAD1 * n_nodes;      // 4N
  float* z1  = m1  + (size_t)NHEAD1 * n_nodes;      // 4N
  float* h2  = z1  + (size_t)NHEAD1 * n_nodes;      // 16N
  float* el2 = h2  + (size_t)NCLASS * n_nodes;      // N
  float* er2 = el2 + (size_t)n_nodes;               // N
  float* m2  = er2 + (size_t)n_nodes;               // N
  float* z2  = m2  + (size_t)n_nodes;               // N

  const int BS = 256;
  auto blocks = [&](long long n) { return dim3((unsigned)((n + BS - 1) / BS)); };

  // ---- init accumulators ----
  k_fill<<<blocks((long long)HD1 * n_nodes), BS, 0, stream>>>(x1, HD1 * n_nodes, 0.f);
  k_fill<<<blocks((long long)NHEAD1 * n_nodes), BS, 0, stream>>>(m1, NHEAD1 * n_nodes, -INFINITY);
  k_fill<<<blocks((long long)NHEAD1 * n_nodes), BS, 0, stream>>>(z1, NHEAD1 * n_nodes, 0.f);
  k_fill<<<blocks(n_nodes), BS, 0, stream>>>(m2, n_nodes, -INFINITY);
  k_fill<<<blocks(n_nodes), BS, 0, stream>>>(z2, n_nodes, 0.f);
  k_fill<<<blocks((long long)NCLASS * n_nodes), BS, 0, stream>>>(out, NCLASS * n_nodes, 0.f);

  // ---- layer 1 ----
  {
    dim3 g((n_nodes + 15) / 16);   // one workgroup per 16-row stripe, 8 waves
    k_gemm_wmma_lds<IN_DIM, HD1 / 16><<<g, 32 * (HD1 / 16), 0, stream>>>(
        features, fc1_w, h1, n_nodes, HD1);
  }
  k_el_er<<<blocks((long long)n_nodes * NHEAD1), BS, 0, stream>>>(
      h1, attn_l1, attn_r1, el1, er1, n_nodes, NHEAD1, DIM1);
  k_edge_max<<<blocks((long long)n_edges * NHEAD1), BS, 0, stream>>>(
      src, dst, el1, er1, m1, n_edges, NHEAD1);
  k_edge_expsum<<<blocks((long long)n_edges * NHEAD1), BS, 0, stream>>>(
      src, dst, el1, er1, m1, z1, n_edges, NHEAD1);
  k_aggregate4<<<blocks((long long)n_edges * (HD1 / 4)), BS, 0, stream>>>(
      src, dst, el1, er1, m1, z1, h1, x1, n_edges, NHEAD1, DIM1);
  k_bias_elu<<<blocks((long long)n_nodes * HD1), BS, 0, stream>>>(
      x1, bias1, n_nodes * HD1, HD1);

  // ---- layer 2 ----
  {
    dim3 g((n_nodes + 15) / 16);   // 1 wave per stripe (Ncols = 16)
    k_gemm_wmma_lds<HD1, NCLASS / 16><<<g, 32 * (NCLASS / 16), 0, stream>>>(
        x1, fc2_w, h2, n_nodes, NCLASS);
  }
  k_el_er<<<blocks(n_nodes), BS, 0, stream>>>(
      h2, attn_l2, attn_r2, el2, er2, n_nodes, 1, NCLASS);
  k_edge_max<<<blocks(n_edges), BS, 0, stream>>>(
      src, dst, el2, er2, m2, n_edges, 1);
  k_edge_expsum<<<blocks(n_edges), BS, 0, stream>>>(
      src, dst, el2, er2, m2, z2, n_edges, 1);
  k_aggregate4<<<blocks((long long)n_edges * (NCLASS / 4)), BS, 0, stream>>>(
      src, dst, el2, er2, m2, z2, h2, out, n_edges, 1, NCLASS);
  k_bias_elu<<<blocks((long long)n_nodes * NCLASS), BS, 0, stream>>>(
      out, bias2, n_nodes * NCLASS, NCLASS);
}